// AttentionModule_24584392802396
// MI455X (gfx1250) — compile-verified
//
#include <hip/hip_runtime.h>
#include <math.h>

// Problem constants (match reference)
constexpr int N_ROWS = 1000000;   // rows of x
constexpr int NFEAT  = 128;       // N2
constexpr int NHID   = 64;        // N0
constexpr int NSEG   = 8192;      // segments
constexpr int GROUPS = N_ROWS / 16;   // 62500 exact 16-row WMMA tiles

typedef __attribute__((ext_vector_type(2))) float v2f;
typedef __attribute__((ext_vector_type(8))) float v8f;

// Order-preserving float<->uint encoding so we can use native u32 atomic max.
__device__ __forceinline__ unsigned enc_f32(float f) {
    unsigned u = __float_as_uint(f);
    return (u & 0x80000000u) ? ~u : (u | 0x80000000u);
}
__device__ __forceinline__ float dec_f32(unsigned e) {
    unsigned u = (e & 0x80000000u) ? (e ^ 0x80000000u) : ~e;
    return __uint_as_float(u);
}
constexpr unsigned ENC_NEG_INF = 0x007FFFFFu;  // enc(-inf)

// ---------------------------------------------------------------------------
// Kernel 0: initialize output (atomically accumulated later) + segment stats
// ---------------------------------------------------------------------------
__global__ void k_init(float* __restrict__ out, unsigned* __restrict__ segmax,
                       float* __restrict__ denom) {
    int t = blockIdx.x * blockDim.x + threadIdx.x;
    if (t < NSEG) { segmax[t] = ENC_NEG_INF; denom[t] = 0.0f; }
    if (t < NSEG * NFEAT) out[t] = 0.0f;
}

// ---------------------------------------------------------------------------
// Kernel 1: scores[i] = (x_i @ W^T + b) . a  via V_WMMA_F32_16X16X4_F32
// One wave handles a 16-row tile; 4 accumulators cover the 64 hidden features.
// A layout (16x4 f32): lane<16 -> row=lane,   K = {k0,  k0+1}
//                      lane>=16 -> row=lane-16, K = {k0+2,k0+3}
// B layout (4x16 f32): lane<16 -> col n=lane,  K = {k0,  k0+1}
//                      lane>=16 -> col n=lane-16, K = {k0+2,k0+3}
// D layout: VGPR r, lanes 0-15 -> D[r][n], lanes 16-31 -> D[8+r][n]
// ---------------------------------------------------------------------------
__global__ __launch_bounds__(256) void
k_scores(const float* __restrict__ x, const int* __restrict__ seg,
         const float* __restrict__ W, const float* __restrict__ bias,
         const float* __restrict__ att, float* __restrict__ scores,
         unsigned* __restrict__ segmax) {
    const int lane  = threadIdx.x & 31;
    const int wave  = threadIdx.x >> 5;
    const int group = blockIdx.x * (blockDim.x >> 5) + wave;

    // c = bias . att  (all 32 lanes end up with the full sum; wave-uniform)
    float c = bias[lane] * att[lane] + bias[lane + 32] * att[lane + 32];
#pragma unroll
    for (int m = 16; m >= 1; m >>= 1) c += __shfl_xor(c, m, 32);

    if (group >= GROUPS) return;   // wave-uniform: EXEC stays all-1s for WMMA

    const int rl   = lane & 15;
    const int koff = (lane >> 4) << 1;

    const v8f zero = {0.f, 0.f, 0.f, 0.f, 0.f, 0.f, 0.f, 0.f};
    v8f acc[4] = {zero, zero, zero, zero};

    const float* xrow = x + (size_t)(group * 16 + rl) * NFEAT + koff;
    const float* w0 = W + (size_t)(0 * 16 + rl) * NFEAT + koff;
    const float* w1 = W + (size_t)(1 * 16 + rl) * NFEAT + koff;
    const float* w2 = W + (size_t)(2 * 16 + rl) * NFEAT + koff;
    const float* w3 = W + (size_t)(3 * 16 + rl) * NFEAT + koff;

#pragma unroll 4
    for (int k0 = 0; k0 < NFEAT; k0 += 4) {
        v2f A = *(const v2f*)(xrow + k0);
        v2f B0 = *(const v2f*)(w0 + k0);
        v2f B1 = *(const v2f*)(w1 + k0);
        v2f B2 = *(const v2f*)(w2 + k0);
        v2f B3 = *(const v2f*)(w3 + k0);
        acc[0] = __builtin_amdgcn_wmma_f32_16x16x4_f32(false, A, false, B0,
                                                       (short)0, acc[0], false, false);
        acc[1] = __builtin_amdgcn_wmma_f32_16x16x4_f32(false, A, false, B1,
                                                       (short)0, acc[1], false, false);
        acc[2] = __builtin_amdgcn_wmma_f32_16x16x4_f32(false, A, false, B2,
                                                       (short)0, acc[2], false, false);
        acc[3] = __builtin_amdgcn_wmma_f32_16x16x4_f32(false, A, false, B3,
                                                       (short)0, acc[3], false, false);
    }

    // Fold hidden dim with attention vector: p[r] = sum_n D[r_or_8+r][n]*a[n]
    float p[8] = {0.f, 0.f, 0.f, 0.f, 0.f, 0.f, 0.f, 0.f};
#pragma unroll
    for (int t = 0; t < 4; ++t) {
        float av = att[t * 16 + rl];
#pragma unroll
        for (int r = 0; r < 8; ++r) p[r] += acc[t][r] * av;
    }
    // Reduce across the 16 lanes of each half-wave (n dimension)
#pragma unroll
    for (int m = 8; m >= 1; m >>= 1) {
#pragma unroll
        for (int r = 0; r < 8; ++r) p[r] += __shfl_xor(p[r], m, 32);
    }

    if (rl == 0) {  // lanes 0 (rows 0..7) and 16 (rows 8..15)
        const int base = group * 16 + (lane >> 4) * 8;
#pragma unroll
        for (int r = 0; r < 8; ++r) {
            const int row = base + r;
            const float s = p[r] + c;
            scores[row] = s;
            atomicMax(&segmax[seg[row]], enc_f32(s));
        }
    }
}

// ---------------------------------------------------------------------------
// Kernel 2: ex = exp(score - seg_max); denom[seg] += ex   (in-place on scores)
// ---------------------------------------------------------------------------
__global__ void k_exp(float* __restrict__ scores, const int* __restrict__ seg,
                      const unsigned* __restrict__ segmax, float* __restrict__ denom) {
    int i = blockIdx.x * blockDim.x + threadIdx.x;
    if (i >= N_ROWS) return;
    const int s = seg[i];
    float m = dec_f32(segmax[s]);
    if (!isfinite(m)) m = 0.0f;  // reference's empty/non-finite guard
    const float e = expf(scores[i] - m);
    scores[i] = e;
    atomicAdd(&denom[s], e);
}

// ---------------------------------------------------------------------------
// Kernel 3: out[seg] += (ex/denom[seg]) * x_row
// One wave per row (strided); lane does float4 of the 128-wide row; the 4 MB
// output is L2-resident so the f32 scatter atomics never hit HBM.
// ---------------------------------------------------------------------------
__global__ __launch_bounds__(256) void
k_accum(const float* __restrict__ x, const int* __restrict__ seg,
        const float* __restrict__ ex, const float* __restrict__ denom,
        float* __restrict__ out) {
    const int lane = threadIdx.x & 31;
    const int gw = (blockIdx.x * blockDim.x + threadIdx.x) >> 5;
    const int nw = (gridDim.x * blockDim.x) >> 5;
    for (int row = gw; row < N_ROWS; row += nw) {
        const int s = seg[row];
        const float w = ex[row] / fmaxf(denom[s], 1e-30f);
        const float4 xv = ((const float4*)(x + (size_t)row * NFEAT))[lane];
        float* o = out + (size_t)s * NFEAT + lane * 4;
        atomicAdd(o + 0, w * xv.x);
        atomicAdd(o + 1, w * xv.y);
        atomicAdd(o + 2, w * xv.z);
        atomicAdd(o + 3, w * xv.w);
    }
}

// ---------------------------------------------------------------------------
extern "C" void kernel_launch(void* const* d_in, const int* in_sizes, int n_in,
                              void* d_out, int out_size, void* d_ws, size_t ws_size,
                              hipStream_t stream) {
    const float* x    = (const float*)d_in[0];
    const int*   seg  = (const int*)d_in[1];
    const float* W    = (const float*)d_in[2];
    const float* bias = (const float*)d_in[3];
    const float* att  = (const float*)d_in[4];
    float* out = (float*)d_out;

    // Workspace layout: scores[N] | segmax[NSEG] (u32) | denom[NSEG]  (~4.07 MB)
    float*    scores = (float*)d_ws;
    unsigned* segmax = (unsigned*)(scores + N_ROWS);
    float*    denom  = (float*)(segmax + NSEG);

    k_init  <<<(NSEG * NFEAT + 255) / 256, 256, 0, stream>>>(out, segmax, denom);
    k_scores<<<(GROUPS + 7) / 8,           256, 0, stream>>>(x, seg, W, bias, att,
                                                             scores, segmax);
    k_exp   <<<(N_ROWS + 255) / 256,       256, 0, stream>>>(scores, seg, segmax, denom);
    k_accum <<<4096,                       256, 0, stream>>>(x, seg, scores, denom, out);
}